// Atten_2576980377664
// MI455X (gfx1250) — compile-verified
//
#include <hip/hip_runtime.h>
#include <hip/hip_bf16.h>
#include <math.h>

typedef __attribute__((ext_vector_type(2))) float v2f;
typedef __attribute__((ext_vector_type(8))) float v8f;

// Problem constants
#define BB 16
#define KK 49
#define TT 256
#define HH 512
#define DD 512

// ---------------------------------------------------------------------------
// GEMM: out[M x 512] = A[M x 512] @ W[512 x 512] + bias (+ addend) (tanh?)
// One wave (32 threads) per 16(M) x 64(N) output tile: 4 f32 WMMA accumulators
// share one A fragment per K-step (4x less A traffic, 4 WMMA per A load).
//
// Fragment layouts (CDNA5 ISA 7.12.2):
//   A 16x4 : lanes 0-15 -> M=lane,   VGPR0=K0, VGPR1=K1
//            lanes16-31 -> M=lane-16, VGPR0=K2, VGPR1=K3
//   B 4x16 : VGPR v: lanes 0-15 -> (K=v,   N=lane)
//                    lanes16-31 -> (K=v+2, N=lane-16)
//   C/D    : VGPR v: lanes 0-15 -> (M=v,   N=lane)
//                    lanes16-31 -> (M=v+8, N=lane-16)
// ---------------------------------------------------------------------------
__global__ void gemm16_wmma(const float* __restrict__ A,
                            const float* __restrict__ W,
                            const float* __restrict__ bias,
                            const float* __restrict__ addend, // nullable, M x 512
                            float* __restrict__ out,
                            int M, int applyTanh) {
    const int lane = threadIdx.x;      // 0..31
    const int half = lane >> 4;        // 0 or 1
    const int l    = lane & 15;
    const int n0   = blockIdx.x * 64;  // 64-wide N tile per wave
    const int m0   = blockIdx.y * 16;

    const float* __restrict__ arow = A + (size_t)(m0 + l) * HH;

    v8f acc0 = {}, acc1 = {}, acc2 = {}, acc3 = {};

    for (int k0 = 0; k0 < HH; k0 += 4) {
        const int ka = k0 + (half << 1);   // K index this half-wave handles

        // prefetch next K-block (A row + 2 W rows) into cache
        if (k0 + 4 < HH) {
            __builtin_prefetch(arow + ka + 4, 0, 3);
            __builtin_prefetch(&W[(size_t)(ka + 4) * DD + n0 + l], 0, 3);
            __builtin_prefetch(&W[(size_t)(ka + 5) * DD + n0 + l], 0, 3);
        }

        v2f a;
        a.x = arow[ka];
        a.y = arow[ka + 1];

        const float* __restrict__ w0 = &W[(size_t)ka * DD + n0 + l];
        const float* __restrict__ w1 = &W[(size_t)(ka + 1) * DD + n0 + l];

        v2f b0, b1, b2, b3;
        b0.x = w0[0];  b0.y = w1[0];
        b1.x = w0[16]; b1.y = w1[16];
        b2.x = w0[32]; b2.y = w1[32];
        b3.x = w0[48]; b3.y = w1[48];

        acc0 = __builtin_amdgcn_wmma_f32_16x16x4_f32(false, a, false, b0,
                                                     (short)0, acc0, false, false);
        acc1 = __builtin_amdgcn_wmma_f32_16x16x4_f32(false, a, false, b1,
                                                     (short)0, acc1, false, false);
        acc2 = __builtin_amdgcn_wmma_f32_16x16x4_f32(false, a, false, b2,
                                                     (short)0, acc2, false, false);
        acc3 = __builtin_amdgcn_wmma_f32_16x16x4_f32(false, a, false, b3,
                                                     (short)0, acc3, false, false);
    }

    // ---- epilogue: bias (+addend) (tanh), store 4 column tiles ------------
    v8f accs[4] = {acc0, acc1, acc2, acc3};
#pragma unroll
    for (int nt = 0; nt < 4; ++nt) {
        const int col = n0 + nt * 16 + l;
        const float bc = bias[col];
#pragma unroll
        for (int v = 0; v < 8; ++v) {
            const int row = m0 + v + half * 8;
            float x = accs[nt][v] + bc;
            if (addend)    x += addend[(size_t)row * DD + col];
            if (applyTanh) x = tanhf(x);
            out[(size_t)row * DD + col] = x;
        }
    }
}

// ---------------------------------------------------------------------------
// z_ext[row] = sum_d tanhU[row,d] * Wh[d] + bh   (row = b*T + t, 4096 rows)
// ---------------------------------------------------------------------------
__global__ void zext_kernel(const float* __restrict__ tanhU,
                            const float* __restrict__ Wh,
                            const float* __restrict__ bh,
                            float* __restrict__ zext) {
    __shared__ float red[256];
    const int row = blockIdx.x;
    const float* __restrict__ u = tanhU + (size_t)row * DD;
    float s = 0.f;
    for (int d = threadIdx.x; d < DD; d += 256)
        s = fmaf(u[d], Wh[d], s);
    red[threadIdx.x] = s;
    __syncthreads();
    for (int off = 128; off > 0; off >>= 1) {
        if ((int)threadIdx.x < off) red[threadIdx.x] += red[threadIdx.x + off];
        __syncthreads();
    }
    if (threadIdx.x == 0) zext[row] = red[0] + bh[0];
}

// ---------------------------------------------------------------------------
// Fused attention: one 256-thread block (8 waves) per (b,t) row.
//   z_t[k]   = sum_d tanh(cv[b,k,d] + cg[row,d]) * Wh[d] + bh     (k < 49)
//   alpha_t  = softmax_49(z_t)
//   beta_t   = softmax_50([z_t, z_ext])[-1]
//   c_t[h]   = sum_k alpha_t[k] * V[b,k,h]
//   c_hat[h] = beta*s_t[row,h] + (1-beta)*c_t[h]
// content_v (411 MB fp32) is never materialized.
// ---------------------------------------------------------------------------
__global__ void attn_kernel(const float* __restrict__ cv,
                            const float* __restrict__ cg,
                            const float* __restrict__ V,
                            const float* __restrict__ s_t,
                            const float* __restrict__ Wh,
                            const float* __restrict__ bh,
                            const float* __restrict__ zext,
                            float* __restrict__ out_chat,
                            float* __restrict__ out_alpha,
                            float* __restrict__ out_beta) {
    __shared__ float cg_s[HH];
    __shared__ float wh_s[HH];
    __shared__ float zbuf[64];
    __shared__ float alpha_s[64];
    __shared__ float beta_sh;

    const int row  = blockIdx.x;        // b*T + t
    const int b    = row >> 8;          // T == 256
    const int tid  = threadIdx.x;
    const int wave = tid >> 5;
    const int lane = tid & 31;

    const float* __restrict__ cgr = cg + (size_t)row * HH;
    for (int d = tid; d < HH; d += 256) {
        cg_s[d] = cgr[d];
        wh_s[d] = Wh[d];
    }
    __syncthreads();

    const float bhv = bh[0];

    // ---- additive attention scores: 8 waves split the 49 keys -------------
    for (int k = wave; k < KK; k += 8) {
        const float* __restrict__ cvr = cv + (size_t)(b * KK + k) * HH;
        // each lane prefetches a distinct 64B line of the NEXT cv row
        if (k + 8 < KK)
            __builtin_prefetch(cvr + (size_t)8 * HH + (lane << 4), 0, 3);
        float s = 0.f;
#pragma unroll 4
        for (int d = lane; d < HH; d += 32)
            s = fmaf(tanhf(cvr[d] + cg_s[d]), wh_s[d], s);
        for (int off = 16; off > 0; off >>= 1)
            s += __shfl_xor(s, off, 32);
        if (lane == 0) zbuf[k] = s + bhv;
    }
    __syncthreads();

    // ---- dual softmax (49-wide and 50-wide), done by wave 0 ---------------
    if (wave == 0) {
        float z0 = (lane < KK)      ? zbuf[lane]      : -INFINITY;
        float z1 = (lane + 32 < KK) ? zbuf[lane + 32] : -INFINITY;
        float mx = fmaxf(z0, z1);
        for (int off = 16; off > 0; off >>= 1)
            mx = fmaxf(mx, __shfl_xor(mx, off, 32));
        float e0 = (lane < KK)      ? expf(z0 - mx) : 0.f;
        float e1 = (lane + 32 < KK) ? expf(z1 - mx) : 0.f;
        float ssum = e0 + e1;
        for (int off = 16; off > 0; off >>= 1)
            ssum += __shfl_xor(ssum, off, 32);
        const float inv = 1.f / ssum;
        if (lane < KK)      alpha_s[lane]      = e0 * inv;
        if (lane + 32 < KK) alpha_s[lane + 32] = e1 * inv;
        if (lane == 0) {
            // softmax over [z_t, z_ext]: reuse (mx, ssum) statistics
            const float ze  = zext[row];
            const float mx2 = fmaxf(mx, ze);
            const float ee  = expf(ze - mx2);
            const float S50 = ssum * expf(mx - mx2) + ee;
            beta_sh = ee / S50;
        }
    }
    __syncthreads();

    if (tid < KK)  out_alpha[(size_t)row * KK + tid] = alpha_s[tid];
    if (tid == 0)  out_beta[row] = beta_sh;

    // ---- c_t = alpha @ V ;  blend with s_t --------------------------------
    const float beta = beta_sh;
    const float* __restrict__ str = s_t + (size_t)row * HH;
    const float* __restrict__ Vb  = V + (size_t)b * KK * HH;
    for (int h = tid; h < HH; h += 256) {
        float acc = 0.f;
#pragma unroll 7
        for (int k = 0; k < KK; ++k)
            acc = fmaf(alpha_s[k], Vb[(size_t)k * HH + h], acc);
        out_chat[(size_t)row * HH + h] = beta * str[h] + (1.f - beta) * acc;
    }
}

// ---------------------------------------------------------------------------
extern "C" void kernel_launch(void* const* d_in, const int* in_sizes, int n_in,
                              void* d_out, int out_size, void* d_ws, size_t ws_size,
                              hipStream_t stream) {
    const float* V   = (const float*)d_in[0];   // (16,49,512)
    const float* h_t = (const float*)d_in[1];   // (16,256,512)
    const float* s_t = (const float*)d_in[2];   // (16,256,512)
    const float* Wv  = (const float*)d_in[3];   // (512,512)
    const float* bv  = (const float*)d_in[4];
    const float* Wg  = (const float*)d_in[5];
    const float* bg  = (const float*)d_in[6];
    const float* Ws  = (const float*)d_in[7];
    const float* bs  = (const float*)d_in[8];
    const float* Wh  = (const float*)d_in[9];   // (512,1)
    const float* bh  = (const float*)d_in[10];  // (1,)

    const int rowsV = BB * KK;   // 784
    const int rows  = BB * TT;   // 4096

    float* ws   = (float*)d_ws;
    float* cvw  = ws;                                   // 784  x 512
    float* cgw  = cvw + (size_t)rowsV * DD;             // 4096 x 512
    float* tuw  = cgw + (size_t)rows * DD;              // 4096 x 512 (tanh(content_s))
    float* zew  = tuw + (size_t)rows * DD;              // 4096

    float* out       = (float*)d_out;
    float* out_chat  = out;                             // 4096 x 512
    float* out_alpha = out_chat + (size_t)rows * HH;    // 4096 x 49
    float* out_beta  = out_alpha + (size_t)rows * KK;   // 4096

    // cv = V @ Wv + bv
    dim3 g1(DD / 64, rowsV / 16);
    gemm16_wmma<<<g1, 32, 0, stream>>>(V, Wv, bv, nullptr, cvw, rowsV, 0);
    // cg = h_t @ Wg + bg
    dim3 g2(DD / 64, rows / 16);
    gemm16_wmma<<<g2, 32, 0, stream>>>(h_t, Wg, bg, nullptr, cgw, rows, 0);
    // tanh(content_s) = tanh(s_t @ Ws + bs + cg)
    gemm16_wmma<<<g2, 32, 0, stream>>>(s_t, Ws, bs, cgw, tuw, rows, 1);
    // z_ext = tanh(content_s) @ Wh + bh
    zext_kernel<<<rows, 256, 0, stream>>>(tuw, Wh, bh, zew);
    // fused attention core
    attn_kernel<<<rows, 256, 0, stream>>>(cvw, cgw, V, s_t, Wh, bh, zew,
                                          out_chat, out_alpha, out_beta);
}